// GraphMaxPooling_82815559402085
// MI455X (gfx1250) — compile-verified
//
#include <hip/hip_runtime.h>
#include <stdint.h>

#define LDS_AS __attribute__((address_space(3)))
#define GLB_AS __attribute__((address_space(1)))

// GCC-vector int4 to match the builtin's prototype ('V4i' = vector_size vector)
typedef int v4i_t __attribute__((vector_size(16)));
typedef GLB_AS v4i_t* glb_v4i_p;
typedef LDS_AS v4i_t* lds_v4i_p;

// Problem constants from the reference
static constexpr int Bn = 64, Cn = 4, Nn = 128, Kn = 64;

// ---------------------------------------------------------------------------
// Async global -> LDS copy of 16 bytes per lane (gfx1250 GLOBAL_LOAD_ASYNC_TO_LDS_B128,
// tracked by ASYNCcnt). Integer round-trips give us legal inttoptr casts into the
// LDS / global address spaces (flat LDS addr low 32 bits == LDS offset).
// ---------------------------------------------------------------------------
__device__ __forceinline__ void async_g2l_b128(const float* g, float* l) {
  const uintptr_t gi = (uintptr_t)g;
  const uint32_t  li = (uint32_t)(uintptr_t)l;
#if __has_builtin(__builtin_amdgcn_global_load_async_to_lds_b128)
  __builtin_amdgcn_global_load_async_to_lds_b128(
      (glb_v4i_p)gi, (lds_v4i_p)(uintptr_t)li, /*offset=*/0, /*cpol=*/0);
#else
  asm volatile("global_load_async_to_lds_b128 %0, %1, off"
               :: "v"(li), "v"(gi) : "memory");
#endif
}

__device__ __forceinline__ void wait_async_zero() {
#if __has_builtin(__builtin_amdgcn_s_wait_asynccnt)
  __builtin_amdgcn_s_wait_asynccnt(0);
#else
  asm volatile("s_wait_asynccnt 0" ::: "memory");
#endif
}

// ---------------------------------------------------------------------------
// One block per batch b. 256 threads = 8 waves (wave32). Stage x[b] (128x64 f32,
// 32 KB) into LDS via async DMA; each wave owns rows i = wave, wave+8, ...;
// each lane owns k = {2*lane, 2*lane+1}. adj row is turned into 4x 32-bit ballot
// masks; the uniform scalar set-bit loop only touches selected x rows.
// ---------------------------------------------------------------------------
__global__ __launch_bounds__(256)
void GraphMaxPooling_kernel(const float* __restrict__ x,
                            const float* __restrict__ adj,
                            float* __restrict__ out) {
  __shared__ float xs[Nn * Kn];  // 32 KB

  const int b    = blockIdx.x;
  const int t    = threadIdx.x;
  const int lane = t & 31;
  const int wave = t >> 5;

  // --- stage x[b] into LDS: 2048 float4 chunks, 8 per thread, async ---
  const float* xb = x + (size_t)b * (Nn * Kn);
  #pragma unroll
  for (int r = 0; r < 8; ++r) {
    const int q = t + r * 256;           // float4 index 0..2047
    async_g2l_b128(xb + q * 4, xs + q * 4);
  }
  wait_async_zero();
  __syncthreads();

  const int kb = 2 * lane;               // this lane's k pair

  for (int i = wave; i < Nn; i += 8) {   // 16 rows per wave, EXEC stays full
    float sum0 = 0.0f, sum1 = 0.0f;

    #pragma unroll
    for (int c = 0; c < Cn; ++c) {
      const float* arow = adj + ((((size_t)b * Cn + c) * Nn) + i) * Nn;
      // prefetch the same channel's row for our next i iteration
      __builtin_prefetch(arow + 8 * Nn, 0, 1);

      float m0 = -__builtin_inff();
      float m1 = -__builtin_inff();
      bool anyZero = false;

      #pragma unroll
      for (int g = 0; g < 4; ++g) {
        const float a = arow[g * 32 + lane];
        uint32_t mask = __builtin_amdgcn_ballot_w32(a != 0.0f);
        mask = (uint32_t)__builtin_amdgcn_readfirstlane((int)mask);  // force SGPR
        anyZero |= (mask != 0xffffffffu);
        const int jbase = g * 32;
        while (mask) {                    // uniform scalar loop over set bits
          const int j = __builtin_ctz(mask);
          mask &= (mask - 1u);
          const float2 v =
              *(const float2*)(xs + (jbase + j) * Kn + kb);  // ds_load_b64, 64-bank clean
          m0 = fmaxf(m0, v.x);
          m1 = fmaxf(m1, v.y);
        }
      }
      // zero entries of adj contribute a 0 candidate to the max
      if (anyZero) { m0 = fmaxf(m0, 0.0f); m1 = fmaxf(m1, 0.0f); }
      sum0 += m0;
      sum1 += m1;
    }

    float2* o = (float2*)(out + ((size_t)b * Nn + i) * Kn);
    o[lane] = make_float2(sum0, sum1);    // global_store_b64, coalesced
  }
}

extern "C" void kernel_launch(void* const* d_in, const int* in_sizes, int n_in,
                              void* d_out, int out_size, void* d_ws, size_t ws_size,
                              hipStream_t stream) {
  (void)in_sizes; (void)n_in; (void)out_size; (void)d_ws; (void)ws_size;
  const float* x   = (const float*)d_in[0];   // (B, N, K) f32
  const float* adj = (const float*)d_in[1];   // (B, C, N, N) f32
  float* out       = (float*)d_out;           // (B, N, K) f32

  GraphMaxPooling_kernel<<<dim3(Bn), dim3(256), 0, stream>>>(x, adj, out);
}